// MV_GCN_29446295781426
// MI455X (gfx1250) — compile-verified
//
#include <hip/hip_runtime.h>

#define B_     256
#define NODES_ 116
#define FEAT_  115
#define LENN_  6670
#define SEG2_  13456
#define HID_   128
#define HC_    512
#define NROWS_ (B_ * 2 * NODES_)   // 59392 rows of h

typedef __attribute__((ext_vector_type(2))) float v2f;
typedef __attribute__((ext_vector_type(8))) float v8f;

// ---------------------------------------------------------------------------
// K1: per (graph, view) GEMM  h[116x128] = x_tile[116x115] @ Wv[115x128]
// FP32 WMMA 16x16x4. 8 waves per WG, wave w owns M-tile w, loops 8 N-tiles.
// Branch-free inner loop: row pointer clamped (OOB M rows are discarded at
// store anyway), K tail peeled so the 28 main k-steps have zero guards.
// ---------------------------------------------------------------------------
__global__ __launch_bounds__(256) void k1_gemm(const float* __restrict__ x,
                                               const float* __restrict__ W1,
                                               const float* __restrict__ W2,
                                               float* __restrict__ h) {
  const int wg = blockIdx.x;                 // 0..511 = g*2 + v
  const int v = wg & 1, g = wg >> 1;
  const float* __restrict__ W = v ? W2 : W1;
  const long base = (long)g * (2 * NODES_) + (long)v * NODES_;
  const int lane = threadIdx.x & 31;
  const int mi   = threadIdx.x >> 5;         // wave id == M tile (0..7)
  const int lr   = lane & 15;
  const int kh   = lane >> 4;                // 0/1: which K-pair half

  v8f acc[8];
  #pragma unroll
  for (int i = 0; i < 8; ++i) acc[i] = (v8f)0.f;

  const int m  = mi * 16 + lr;
  const int mc = (m < NODES_) ? m : (NODES_ - 1);   // clamp: rows >=116 discarded
  const float* __restrict__ xr = x + (base + mc) * FEAT_;

  // Main K loop: k0 = 0,4,...,108  (columns touched: <= 111 < 115, no guards)
  for (int k0 = 0; k0 < 112; k0 += 4) {
    const int ka = k0 + 2 * kh;
    v2f a; a.x = xr[ka]; a.y = xr[ka + 1];
    const float* __restrict__ w0 = W + ka * HID_;
    #pragma unroll
    for (int ni = 0; ni < 8; ++ni) {
      const int n = ni * 16 + lr;
      v2f b; b.x = w0[n]; b.y = w0[HID_ + n];
      acc[ni] = __builtin_amdgcn_wmma_f32_16x16x4_f32(
          false, a, false, b, (short)0, acc[ni], false, false);
    }
  }
  // Peeled tail k0 = 112: kh=0 lanes cover cols 112,113; kh=1 cover 114,(115->0)
  {
    const int ka = 112 + 2 * kh;
    v2f a; a.x = xr[ka];
    a.y = kh ? 0.f : xr[113];                       // branchless cndmask
    const float* __restrict__ w0 = W + ka * HID_;
    const float* __restrict__ w1 = W + 113 * HID_;
    #pragma unroll
    for (int ni = 0; ni < 8; ++ni) {
      const int n = ni * 16 + lr;
      v2f b; b.x = w0[n];
      b.y = kh ? 0.f : w1[n];
      acc[ni] = __builtin_amdgcn_wmma_f32_16x16x4_f32(
          false, a, false, b, (short)0, acc[ni], false, false);
    }
  }

  // D layout: VGPR r -> rows r (lanes 0-15) and r+8 (lanes 16-31), N = lane.low
  float* __restrict__ hb = h + (base + mi * 16 + 8 * kh) * HID_ + lr;
  if (mi < 7) {                                     // tiles 0..6 fully in range
    #pragma unroll
    for (int ni = 0; ni < 8; ++ni)
      #pragma unroll
      for (int r = 0; r < 8; ++r)
        hb[(long)r * HID_ + ni * 16] = acc[ni][r];
  } else {                                          // tile 7: rows 112..127
    #pragma unroll
    for (int ni = 0; ni < 8; ++ni)
      #pragma unroll
      for (int r = 0; r < 8; ++r)
        if (112 + r + 8 * kh < NODES_)
          hb[(long)r * HID_ + ni * 16] = acc[ni][r];
  }
}

// ---------------------------------------------------------------------------
// K2: per (graph, view) weighted-edge aggregation + self-loop + bias + chan-max
// acc[116x128] in LDS; channel c = lane + 32k => conflict-free ds_add_f32
// (row stride 128 floats == 0 mod 64 banks, bank depends only on c).
// Edge (src,dst,norm) broadcast via v_readlane -> scalar row bases.
// ---------------------------------------------------------------------------
__global__ __launch_bounds__(256) void k2_agg(const int*   __restrict__ ei,
                                              const float* __restrict__ ew,
                                              const float* __restrict__ h,
                                              const float* __restrict__ b1,
                                              const float* __restrict__ b2,
                                              float* __restrict__ feat) {
  __shared__ float acc[NODES_ * HID_];   // 59392 B
  __shared__ float dinv[NODES_];

  const int wg = blockIdx.x;
  const int v = wg & 1, g = wg >> 1;
  const int tid = threadIdx.x;
  const int lane = tid & 31, wave = tid >> 5;
  const long ebase = (long)g * SEG2_ + (long)v * LENN_;
  const int  noff  = g * (2 * NODES_) + v * NODES_;
  const long hbase = (long)noff * HID_;
  const int* __restrict__ src = ei;
  const int* __restrict__ dst = ei + (long)B_ * SEG2_;
  const float* __restrict__ bias = v ? b2 : b1;

  for (int i = tid; i < NODES_ * HID_; i += 256) acc[i] = 0.f;
  if (tid < NODES_) dinv[tid] = 0.f;
  __syncthreads();

  // weighted in-degree
  for (int e = tid; e < LENN_; e += 256) {
    const int d = dst[ebase + e] - noff;
    atomicAdd(&dinv[d], ew[ebase + e]);
  }
  __syncthreads();
  if (tid < NODES_) dinv[tid] = rsqrtf(dinv[tid] + 1.0f);  // +1 self-loop
  __syncthreads();

  // edge scatter: each wave batches 32 edges (coalesced), broadcasts via readlane
  for (int e0 = wave * 32; e0 < LENN_; e0 += 256) {
    const int e = e0 + lane;
    int s = 0, d = 0; float nrm = 0.f;
    if (e < LENN_) {
      s = src[ebase + e] - noff;
      d = dst[ebase + e] - noff;
      nrm = dinv[s] * ew[ebase + e] * dinv[d];
    }
    const int cnt = min(32, LENN_ - e0);
    for (int i = 0; i < cnt; ++i) {
      const int   si = __builtin_amdgcn_readlane(s, i);               // SGPR
      const int   di = __builtin_amdgcn_readlane(d, i);               // SGPR
      const float nv =
          __int_as_float(__builtin_amdgcn_readlane(__float_as_int(nrm), i));
      const float* __restrict__ hr = h + hbase + (long)si * HID_;     // scalar base
      float* __restrict__ ar = acc + di * HID_;                       // scalar base
      #pragma unroll
      for (int k = 0; k < 4; ++k) {
        const int c = lane + 32 * k;                 // 32 distinct LDS banks
        atomicAdd(&ar[c], nv * hr[c]);
      }
    }
  }
  __syncthreads();

  // self-loop + bias + max over channels; feat[b, 2n+v] interleaved layout
  for (int n = wave; n < NODES_; n += 8) {
    const float di2 = dinv[n] * dinv[n];
    const float* __restrict__ hr = h + hbase + (long)n * HID_;
    float m = -3.402823466e38f;
    #pragma unroll
    for (int k = 0; k < 4; ++k) {
      const int c = lane + 32 * k;
      m = fmaxf(m, acc[n * HID_ + c] + di2 * hr[c] + bias[c]);
    }
    #pragma unroll
    for (int off = 16; off > 0; off >>= 1)
      m = fmaxf(m, __shfl_xor(m, off, 32));
    if (lane == 0) feat[g * (2 * NODES_) + 2 * n + v] = m;
  }
}

// ---------------------------------------------------------------------------
// K3a: h6 = relu(feat[256x232] @ W6[232x512] + b6)
// ---------------------------------------------------------------------------
__global__ __launch_bounds__(256) void k3_fc1(const float* __restrict__ feat,
                                              const float* __restrict__ W6,
                                              const float* __restrict__ b6,
                                              float* __restrict__ h6) {
  const int idx = blockIdx.x * 256 + threadIdx.x;
  if (idx >= B_ * HC_) return;
  const int b = idx >> 9;        // /512
  const int j = idx & (HC_ - 1);
  const float* __restrict__ fr = feat + b * (2 * NODES_);
  float sum = b6[j];
  for (int k = 0; k < 2 * NODES_; ++k) sum += fr[k] * W6[k * HC_ + j];
  h6[idx] = fmaxf(sum, 0.f);
}

// ---------------------------------------------------------------------------
// K3b: out = h6[256x512] @ W7[512x1] + b7
// ---------------------------------------------------------------------------
__global__ __launch_bounds__(256) void k3_fc2(const float* __restrict__ h6,
                                              const float* __restrict__ W7,
                                              const float* __restrict__ b7,
                                              float* __restrict__ out) {
  const int b = blockIdx.x * blockDim.x + threadIdx.x;
  if (b >= B_) return;
  const float* __restrict__ hr = h6 + b * HC_;
  float sum = b7[0];
  for (int j = 0; j < HC_; ++j) sum += hr[j] * W7[j];
  out[b] = sum;
}

// ---------------------------------------------------------------------------
extern "C" void kernel_launch(void* const* d_in, const int* in_sizes, int n_in,
                              void* d_out, int out_size, void* d_ws, size_t ws_size,
                              hipStream_t stream) {
  const float* x  = (const float*)d_in[0];
  const int*   ei = (const int*)  d_in[1];
  const float* ew = (const float*)d_in[2];
  // d_in[3] = batch (unused)
  const float* W1 = (const float*)d_in[4];
  const float* b1 = (const float*)d_in[5];
  const float* W2 = (const float*)d_in[6];
  const float* b2 = (const float*)d_in[7];
  const float* W6 = (const float*)d_in[8];
  const float* b6 = (const float*)d_in[9];
  const float* W7 = (const float*)d_in[10];
  const float* b7 = (const float*)d_in[11];
  float* out = (float*)d_out;

  float* h    = (float*)d_ws;                       // [59392 x 128]
  float* feat = h    + (size_t)NROWS_ * HID_;       // [256 x 232]
  float* h6   = feat + (size_t)B_ * 2 * NODES_;     // [256 x 512]

  k1_gemm<<<B_ * 2, 256, 0, stream>>>(x, W1, W2, h);
  k2_agg <<<B_ * 2, 256, 0, stream>>>(ei, ew, h, b1, b2, feat);
  k3_fc1 <<<(B_ * HC_ + 255) / 256, 256, 0, stream>>>(feat, W6, b6, h6);
  k3_fc2 <<<1, 256, 0, stream>>>(h6, W7, b7, out);
}